// Group_29368986370388
// MI455X (gfx1250) — compile-verified
//
#include <hip/hip_runtime.h>

// Problem constants (match reference)
#define B_   32
#define N_   8192
#define G_   512
#define K_   32
#define CAP  256     // candidate capacity per row (>= 32 + boundary-bucket slack)
#define NBKT 4096    // radix buckets: float bits >> 20 (sign=0 after clamp)

typedef __attribute__((ext_vector_type(2))) float v2f;
typedef __attribute__((ext_vector_type(8))) float v8f;

// ---------------------------------------------------------------------------
// Kernel 1: Farthest Point Sampling. One block per batch, 1024 threads.
// Points cached in LDS (SoA), running min-dists in registers (8/thread),
// two-level wave32 shuffle argmax per step. Writes center (B,G,3) to d_out.
// ---------------------------------------------------------------------------
__global__ void fps_kernel(const float* __restrict__ pts,
                           float* __restrict__ center_out) {
  extern __shared__ float smem[];
  float* xs   = smem;              // N_
  float* ys   = xs + N_;           // N_
  float* zs   = ys + N_;           // N_
  float* redv = zs + N_;           // 32
  int*   redi = (int*)(redv + 32); // 32
  float* bc   = (float*)(redi + 32); // 4 (centroid x,y,z)

  const int b    = blockIdx.x;
  const int tid  = threadIdx.x;
  const int lane = tid & 31;
  const int wid  = tid >> 5;
  const float* p = pts + (size_t)b * N_ * 3;

  for (int k = tid; k < N_ * 3; k += 1024) {
    float v = p[k];
    int n = k / 3, c = k - n * 3;
    if (c == 0) xs[n] = v; else if (c == 1) ys[n] = v; else zs[n] = v;
  }
  if (tid == 0) { bc[0] = p[0]; bc[1] = p[1]; bc[2] = p[2]; }
  __syncthreads();

  float dist[8];
#pragma unroll
  for (int j = 0; j < 8; ++j) dist[j] = 1e10f;

  float* cout = center_out + (size_t)b * G_ * 3;

  for (int g = 0; g < G_; ++g) {
    float fx = bc[0], fy = bc[1], fz = bc[2];
    if (tid == 0) { cout[g*3+0] = fx; cout[g*3+1] = fy; cout[g*3+2] = fz; }

    float bv = -1.0f; int bi = 0;
#pragma unroll
    for (int j = 0; j < 8; ++j) {
      int n = tid + j * 1024;            // SoA + stride-1024: bank-conflict-free
      float dx = xs[n] - fx, dy = ys[n] - fy, dz = zs[n] - fz;
      float dd = dx*dx + dy*dy + dz*dz;
      float dm = fminf(dist[j], dd);
      dist[j] = dm;
      if (dm > bv || (dm == bv && n < bi)) { bv = dm; bi = n; }
    }
    // wave32 argmax (first-max tie-break: smaller index)
#pragma unroll
    for (int off = 16; off > 0; off >>= 1) {
      float ov = __shfl_xor(bv, off, 32);
      int   oi = __shfl_xor(bi, off, 32);
      if (ov > bv || (ov == bv && oi < bi)) { bv = ov; bi = oi; }
    }
    if (lane == 0) { redv[wid] = bv; redi[wid] = bi; }
    __syncthreads();
    if (wid == 0) {
      float v = redv[lane]; int i = redi[lane];
#pragma unroll
      for (int off = 16; off > 0; off >>= 1) {
        float ov = __shfl_xor(v, off, 32);
        int   oi = __shfl_xor(i, off, 32);
        if (ov > v || (ov == v && oi < i)) { v = ov; i = oi; }
      }
      if (lane == 0) { bc[0] = xs[i]; bc[1] = ys[i]; bc[2] = zs[i]; }
    }
    __syncthreads();
  }
}

// ---------------------------------------------------------------------------
// Kernel 2: WMMA-based KNN (radix-select) + gather. One block per
// (batch, 16-center tile): grid = 32*32, block = 256 (8 waves).
// d2 tiles come from V_WMMA_F32_16X16X4_F32:
//   A row m = (-2cx,-2cy,-2cz, |c|^2), B col n = (px,py,pz,1), + |p|^2.
// ---------------------------------------------------------------------------
__device__ __forceinline__ v8f d2_tile(v2f A, const float* __restrict__ p,
                                       int n, bool loHalf, float& sqp_out) {
  float px = p[n*3+0], py = p[n*3+1], pz = p[n*3+2];
  sqp_out = px*px + py*py + pz*pz;
  v2f Bv;
  if (loHalf) { Bv[0] = px; Bv[1] = py; }   // VGPR0=K0, VGPR1=K1 (lanes 0-15)
  else        { Bv[0] = pz; Bv[1] = 1.0f; } // VGPR0=K2, VGPR1=K3 (lanes 16-31)
  v8f acc = {};
  acc = __builtin_amdgcn_wmma_f32_16x16x4_f32(false, A, false, Bv,
                                              (short)0, acc, false, false);
  return acc;
}

__global__ void knn_kernel(const float* __restrict__ pts,
                           const float* __restrict__ centers,
                           float* __restrict__ nbh_out) {
  extern __shared__ float smemf[];
  unsigned* hist   = (unsigned*)smemf;        // 16*NBKT
  unsigned* candHi = hist + 16*NBKT;          // 16*CAP (d2 bits)
  unsigned* candLo = candHi + 16*CAP;         // 16*CAP (point idx)
  float*    cen    = (float*)(candLo + 16*CAP); // 16*4 (cx,cy,cz,|c|^2)
  unsigned* cnt    = (unsigned*)(cen + 64);   // 16
  unsigned* selT   = cnt + 16;                // 16
  unsigned* part   = selT + 16;               // 256

  const int tid  = threadIdx.x;
  const int lane = tid & 31;
  const int wid  = tid >> 5;
  const int b    = blockIdx.x >> 5;  // / 32
  const int gt   = blockIdx.x & 31;
  const float* p = pts + (size_t)b * N_ * 3;
  const float* c = centers + ((size_t)b * G_ + (size_t)gt * 16) * 3;

  for (int i = tid; i < 16*NBKT; i += 256) hist[i] = 0u;
  if (tid < 16) {
    float cx = c[tid*3], cy = c[tid*3+1], cz = c[tid*3+2];
    cen[tid*4+0] = cx; cen[tid*4+1] = cy; cen[tid*4+2] = cz;
    cen[tid*4+3] = cx*cx + cy*cy + cz*cz;
    cnt[tid] = 0u;
  }
  __syncthreads();

  // Per-lane A operand (constant over all chunks). ISA 16x4 F32 A layout:
  // lanes 0-15: VGPR0=K0, VGPR1=K1; lanes 16-31: VGPR0=K2, VGPR1=K3.
  const int m = lane & 15;
  const bool loHalf = (lane < 16);
  v2f A;
  if (loHalf) { A[0] = -2.0f*cen[m*4+0]; A[1] = -2.0f*cen[m*4+1]; }
  else        { A[0] = -2.0f*cen[m*4+2]; A[1] = cen[m*4+3]; }
  const int rowOff = loHalf ? 0 : 8;

  // ---- Pass 1: per-row histogram of d2 bit buckets ----
  for (int chunk = wid; chunk < N_/16; chunk += 8) {
    int n = chunk*16 + (lane & 15);
    float sqp;
    v8f acc = d2_tile(A, p, n, loHalf, sqp);
#pragma unroll
    for (int v = 0; v < 8; ++v) {
      float d2 = fmaxf(acc[v] + sqp, 0.0f);
      unsigned bkt = __float_as_uint(d2) >> 20;
      atomicAdd(&hist[(v + rowOff)*NBKT + bkt], 1u);
    }
  }
  __syncthreads();

  // ---- Find rank-32 bucket per row ----
  {
    int row = tid >> 4, sub = tid & 15;
    unsigned s = 0;
    for (int i = 0; i < 256; ++i) s += hist[row*NBKT + sub*256 + i];
    part[tid] = s;
  }
  __syncthreads();
  if (tid < 16) {
    unsigned cum = 0; int seg = 0;
    for (; seg < 15; ++seg) {
      unsigned ps = part[tid*16 + seg];
      if (cum + ps >= (unsigned)K_) break;
      cum += ps;
    }
    unsigned T = (unsigned)(seg*256 + 255);
    for (int i = 0; i < 256; ++i) {
      cum += hist[tid*NBKT + seg*256 + i];
      if (cum >= (unsigned)K_) { T = (unsigned)(seg*256 + i); break; }
    }
    selT[tid] = T;
  }
  __syncthreads();

  unsigned tloc[8];
#pragma unroll
  for (int v = 0; v < 8; ++v) tloc[v] = selT[v + rowOff];

  // ---- Pass 2: recompute d2 (WMMA) and collect candidates <= threshold ----
  for (int chunk = wid; chunk < N_/16; chunk += 8) {
    int n = chunk*16 + (lane & 15);
    float sqp;
    v8f acc = d2_tile(A, p, n, loHalf, sqp);
#pragma unroll
    for (int v = 0; v < 8; ++v) {
      float d2 = fmaxf(acc[v] + sqp, 0.0f);
      unsigned bkt = __float_as_uint(d2) >> 20;
      int row = v + rowOff;
      if (bkt <= tloc[v]) {
        unsigned pos = atomicAdd(&cnt[row], 1u);
        if (pos < CAP) {
          candHi[row*CAP + pos] = __float_as_uint(d2);
          candLo[row*CAP + pos] = (unsigned)n;
        }
      }
    }
  }
  __syncthreads();

  // ---- Select 32 smallest (d2,idx) keys per row, sorted; write output ----
  for (int rr = 0; rr < 2; ++rr) {
    int row = wid + rr*8;
    int g = gt*16 + row;
    unsigned M = cnt[row]; if (M > CAP) M = CAP;
    float cx = cen[row*4+0], cy = cen[row*4+1], cz = cen[row*4+2];

    unsigned long long key[8];
#pragma unroll
    for (int s = 0; s < 8; ++s) {
      int i = lane + s*32;
      key[s] = (i < (int)M)
        ? ((((unsigned long long)candHi[row*CAP + i]) << 32) | candLo[row*CAP + i])
        : ~0ull;
    }

    float* out = nbh_out + (((size_t)b * G_ + g) * K_) * 3;
    for (int k = 0; k < K_; ++k) {
      unsigned long long bkey = key[0]; int slot = 0;
#pragma unroll
      for (int s = 1; s < 8; ++s)
        if (key[s] < bkey) { bkey = key[s]; slot = s; }
      int meta = (lane << 3) | slot;
#pragma unroll
      for (int off = 16; off > 0; off >>= 1) {
        unsigned ohi = __shfl_xor((unsigned)(bkey >> 32), off, 32);
        unsigned olo = __shfl_xor((unsigned)(bkey & 0xffffffffu), off, 32);
        int      om  = __shfl_xor(meta, off, 32);
        unsigned long long okey = (((unsigned long long)ohi) << 32) | olo;
        if (okey < bkey) { bkey = okey; meta = om; }
      }
      // winner lane invalidates its slot (static indexing, no spill)
#pragma unroll
      for (int s = 0; s < 8; ++s)
        if (meta == ((lane << 3) | s)) key[s] = ~0ull;
      if (lane == 0) {
        unsigned n = (unsigned)(bkey & 0xffffffffu);
        out[k*3+0] = p[n*3+0] - cx;
        out[k*3+1] = p[n*3+1] - cy;
        out[k*3+2] = p[n*3+2] - cz;
      }
    }
  }
}

// ---------------------------------------------------------------------------
extern "C" void kernel_launch(void* const* d_in, const int* in_sizes, int n_in,
                              void* d_out, int out_size, void* d_ws, size_t ws_size,
                              hipStream_t stream) {
  (void)in_sizes; (void)n_in; (void)out_size; (void)d_ws; (void)ws_size;
  const float* pts = (const float*)d_in[0];
  float* out = (float*)d_out;
  float* nbh = out;                                  // (B,G,K,3)
  float* cen = out + (size_t)B_ * G_ * K_ * 3;       // (B,G,3)

  size_t sh1 = (size_t)(3*N_ + 32 + 32 + 4) * 4;     // ~98.6 KB
  fps_kernel<<<dim3(B_), dim3(1024), sh1, stream>>>(pts, cen);

  size_t sh2 = (size_t)(16*NBKT + 2*16*CAP + 64 + 16 + 16 + 256) * 4; // ~296 KB
  knn_kernel<<<dim3(B_ * (G_/16)), dim3(256), sh2, stream>>>(pts, cen, nbh);
}